// GCN_82394652606746
// MI455X (gfx1250) — compile-verified
//
#include <hip/hip_runtime.h>
#include <hip/hip_bf16.h>

typedef __attribute__((ext_vector_type(2))) float v2f;
typedef __attribute__((ext_vector_type(8))) float v8f;

#define TPB 256

// ---------------- degree / norm ----------------

__global__ void deg_init_kernel(float* deg, int n) {
    int i = blockIdx.x * blockDim.x + threadIdx.x;
    if (i < n) deg[i] = 1.0f;               // self-loop contributes 1
}

__global__ void deg_accum_kernel(const int* __restrict__ dst, float* deg, int e) {
    int i = blockIdx.x * blockDim.x + threadIdx.x;
    if (i < e) unsafeAtomicAdd(&deg[dst[i]], 1.0f);
}

__global__ void deg_to_dinv_kernel(float* deg, int n) {
    int i = blockIdx.x * blockDim.x + threadIdx.x;
    if (i < n) deg[i] = rsqrtf(deg[i]);     // deg >= 1 always
}

// ---------------- aggregation: agg = A_norm @ feat (input-dim space) ----------------

template <int D>
__global__ void self_init_kernel(const float* __restrict__ feat,
                                 const float* __restrict__ dinv,
                                 float* __restrict__ agg, int n) {
    int i = blockIdx.x * blockDim.x + threadIdx.x;
    if (i >= n) return;
    float nrm = dinv[i] * dinv[i];
    const float4* fp = (const float4*)(feat + (size_t)i * D);
    float4* ap = (float4*)(agg + (size_t)i * D);
    #pragma unroll
    for (int q = 0; q < D / 4; ++q) {
        float4 v = fp[q];
        v.x *= nrm; v.y *= nrm; v.z *= nrm; v.w *= nrm;
        ap[q] = v;
    }
}

template <int D>
__global__ void edge_agg_kernel(const int* __restrict__ src, const int* __restrict__ dst,
                                const float* __restrict__ dinv,
                                const float* __restrict__ feat,
                                float* __restrict__ agg, int e) {
    int i = blockIdx.x * blockDim.x + threadIdx.x;
    if (i >= e) return;
    int s = src[i], d = dst[i];
    float nrm = dinv[s] * dinv[d];
    const float4* fp = (const float4*)(feat + (size_t)s * D);
    float* ad = agg + (size_t)d * D;
    #pragma unroll
    for (int q = 0; q < D / 4; ++q) {
        float4 v = fp[q];
        unsafeAtomicAdd(ad + 4 * q + 0, v.x * nrm);
        unsafeAtomicAdd(ad + 4 * q + 1, v.y * nrm);
        unsafeAtomicAdd(ad + 4 * q + 2, v.z * nrm);
        unsafeAtomicAdd(ad + 4 * q + 3, v.w * nrm);
    }
}

// ---------------- WMMA fp32 GEMM + bias + ReLU ----------------
// out[N, 16*NT] = relu(A[N, K] @ W[K, 16*NT] + bias), one wave per 16-row group.
// Uses V_WMMA_F32_16X16X4_F32; K % 4 == 0.

template <int K, int NT>
__global__ void gemm_relu_kernel(const float* __restrict__ A,
                                 const float* __restrict__ W,
                                 const float* __restrict__ bias,
                                 float* __restrict__ out, int nrowgrp) {
    const int NTOT = NT * 16;
    int wave = blockIdx.x * (blockDim.x >> 5) + (threadIdx.x >> 5);
    if (wave >= nrowgrp) return;                    // wave-uniform: EXEC all-1s below
    int lane  = threadIdx.x & 31;
    int m     = lane & 15;
    int khalf = (lane >> 4) << 1;                   // 0 (lanes 0-15) or 2 (lanes 16-31)
    int row_base = wave * 16;

    // A fragments: 16x4 f32 per WMMA -> v2f per lane (K=khalf, khalf+1 of the k-block)
    v2f afrag[K / 4];
    #pragma unroll
    for (int kb = 0; kb < K / 4; ++kb) {
        const float* ap = A + (size_t)(row_base + m) * K + kb * 4 + khalf;
        afrag[kb].x = ap[0];
        afrag[kb].y = ap[1];
    }

    #pragma unroll
    for (int t = 0; t < NT; ++t) {
        int n = t * 16 + (lane & 15);
        v8f acc = {};
        #pragma unroll
        for (int kb = 0; kb < K / 4; ++kb) {
            int kk = kb * 4 + khalf;
            v2f bfrag;
            bfrag.x = W[(size_t)kk * NTOT + n];
            bfrag.y = W[(size_t)(kk + 1) * NTOT + n];
            acc = __builtin_amdgcn_wmma_f32_16x16x4_f32(
                /*neg_a=*/false, afrag[kb], /*neg_b=*/false, bfrag,
                /*c_mod=*/(short)0, acc, /*reuse_a=*/false, /*reuse_b=*/false);
        }
        float bv = bias[n];
        int rbase = (lane >> 4) << 3;               // D rows: +0 (lanes 0-15) or +8
        #pragma unroll
        for (int r = 0; r < 8; ++r) {
            float v = acc[r] + bv;
            v = v > 0.0f ? v : 0.0f;
            out[(size_t)(row_base + rbase + r) * NTOT + n] = v;
        }
    }
}

// scalar tail for rows not covered by full 16-row groups (N % 16, normally 0)
template <int K, int NT>
__global__ void gemm_relu_tail_kernel(const float* __restrict__ A,
                                      const float* __restrict__ W,
                                      const float* __restrict__ bias,
                                      float* __restrict__ out,
                                      int row_start, int nrows) {
    const int NTOT = NT * 16;
    int idx = blockIdx.x * blockDim.x + threadIdx.x;
    int r = idx / NTOT, c = idx % NTOT;
    if (r >= nrows) return;
    int row = row_start + r;
    float acc = bias[c];
    for (int k = 0; k < K; ++k) acc += A[(size_t)row * K + k] * W[(size_t)k * NTOT + c];
    out[(size_t)row * NTOT + c] = acc > 0.0f ? acc : 0.0f;
}

// ---------------- head: sigmoid(pc1 * (h2 . Wfc) + bfc) ----------------

__global__ void head_kernel(const float* __restrict__ h2,
                            const float* __restrict__ pc1,
                            const float* __restrict__ Wfc,
                            const float* __restrict__ bfc,
                            float* __restrict__ out, int n) {
    int i = blockIdx.x * blockDim.x + threadIdx.x;
    if (i >= n) return;
    const float4* hp = (const float4*)(h2 + (size_t)i * 64);
    const float4* wp = (const float4*)Wfc;
    float acc = 0.0f;
    #pragma unroll
    for (int q = 0; q < 16; ++q) {
        float4 h = hp[q], w = wp[q];
        acc += h.x * w.x + h.y * w.y + h.z * w.z + h.w * w.w;
    }
    float z = pc1[i] * acc + bfc[0];
    out[i] = 1.0f / (1.0f + __expf(-z));
}

// ---------------- launcher ----------------

extern "C" void kernel_launch(void* const* d_in, const int* in_sizes, int n_in,
                              void* d_out, int out_size, void* d_ws, size_t ws_size,
                              hipStream_t stream) {
    const float* x   = (const float*)d_in[0];   // [N,16]
    const int*   ei  = (const int*)d_in[1];     // [2,E]
    const float* pc1 = (const float*)d_in[2];   // [N,1]
    const float* W1  = (const float*)d_in[3];   // [16,32]
    const float* b1  = (const float*)d_in[4];   // [32]
    const float* W2  = (const float*)d_in[5];   // [32,64]
    const float* b2  = (const float*)d_in[6];   // [64]
    const float* Wfc = (const float*)d_in[7];   // [64,1]
    const float* bfc = (const float*)d_in[8];   // [1]
    float* out = (float*)d_out;

    const int N = in_sizes[0] / 16;
    const int E = in_sizes[1] / 2;
    const int* src = ei;
    const int* dst = ei + E;

    // scratch layout (floats): dinv N | agg1 N*16 | h1 N*32 | agg2 N*32 | h2 N*64
    float* dinv = (float*)d_ws;
    float* agg1 = dinv + N;
    float* h1   = agg1 + (size_t)N * 16;
    float* agg2 = h1   + (size_t)N * 32;
    float* h2   = agg2 + (size_t)N * 32;

    const int gN = (N + TPB - 1) / TPB;
    const int gE = (E + TPB - 1) / TPB;

    // 1) symmetric degree normalization (self-loops included)
    deg_init_kernel<<<gN, TPB, 0, stream>>>(dinv, N);
    deg_accum_kernel<<<gE, TPB, 0, stream>>>(dst, dinv, E);
    deg_to_dinv_kernel<<<gN, TPB, 0, stream>>>(dinv, N);

    // 2) layer-1 aggregation in input space (d=16), then WMMA transform
    self_init_kernel<16><<<gN, TPB, 0, stream>>>(x, dinv, agg1, N);
    edge_agg_kernel<16><<<gE, TPB, 0, stream>>>(src, dst, dinv, x, agg1, E);

    const int nrg = N / 16;
    const int rem = N % 16;
    if (nrg > 0) {
        int blocks = (nrg + 7) / 8;                 // 8 waves per 256-thread block
        gemm_relu_kernel<16, 2><<<blocks, TPB, 0, stream>>>(agg1, W1, b1, h1, nrg);
    }
    if (rem > 0) {
        int tail = rem * 32;
        gemm_relu_tail_kernel<16, 2><<<(tail + TPB - 1) / TPB, TPB, 0, stream>>>(
            agg1, W1, b1, h1, nrg * 16, rem);
    }

    // 3) layer-2 aggregation in 32-dim space, then WMMA transform
    self_init_kernel<32><<<gN, TPB, 0, stream>>>(h1, dinv, agg2, N);
    edge_agg_kernel<32><<<gE, TPB, 0, stream>>>(src, dst, dinv, h1, agg2, E);

    if (nrg > 0) {
        int blocks = (nrg + 7) / 8;
        gemm_relu_kernel<32, 4><<<blocks, TPB, 0, stream>>>(agg2, W2, b2, h2, nrg);
    }
    if (rem > 0) {
        int tail = rem * 64;
        gemm_relu_tail_kernel<32, 4><<<(tail + TPB - 1) / TPB, TPB, 0, stream>>>(
            agg2, W2, b2, h2, nrg * 16, rem);
    }

    // 4) gated head + sigmoid
    head_kernel<<<gN, TPB, 0, stream>>>(h2, pc1, Wfc, bfc, out, N);
}